// RGCNLayer_36988258353760
// MI455X (gfx1250) — compile-verified
//
#include <hip/hip_runtime.h>
#include <hip/hip_bf16.h>
#include <stdint.h>

typedef __attribute__((ext_vector_type(2))) float v2f;
typedef __attribute__((ext_vector_type(8))) float v8f;

static constexpr int IN  = 256;
static constexpr int OUT = 256;
static constexpr int TM  = 32;   // rows (edges/nodes) per WMMA tile block (2 M-tiles)
static constexpr int LDP = IN + 4;  // padded LDS row stride (1040 B: 16B-aligned, bank-skewed)

// ---------------- kernel: zero workspace pieces ----------------
__global__ void k_zero(float* agg, float* deg, int* counts, int N, int R) {
  long long total = (long long)N * OUT + N + R;
  long long stride = (long long)gridDim.x * blockDim.x;
  for (long long i = (long long)blockIdx.x * blockDim.x + threadIdx.x;
       i < total; i += stride) {
    if (i < (long long)N * OUT)            agg[i] = 0.0f;
    else if (i < (long long)N * OUT + N)   deg[i - (long long)N * OUT] = 0.0f;
    else                                   counts[i - (long long)N * OUT - N] = 0;
  }
}

// ---------------- kernel: W[r] = coef[r] @ basis  ([R,IN,OUT]) ----------------
__global__ void k_weights(const float* __restrict__ coef,
                          const float* __restrict__ basis,
                          float* __restrict__ W, int B) {
  __shared__ float c[64];
  int r = blockIdx.y;
  if ((int)threadIdx.x < B) c[threadIdx.x] = coef[(size_t)r * B + threadIdx.x];
  __syncthreads();
  float* wr = W + (size_t)r * IN * OUT;
  int base = blockIdx.x * 1024;
#pragma unroll
  for (int i = 0; i < 4; ++i) {
    int idx = base + i * 256 + threadIdx.x;
    float acc = 0.0f;
    for (int b = 0; b < B; ++b)
      acc += c[b] * basis[(size_t)b * IN * OUT + idx];
    wr[idx] = acc;
  }
}

// ---------------- kernel: per-relation histogram ----------------
__global__ void k_hist(const int* __restrict__ etype, int* counts, int E) {
  int e = blockIdx.x * blockDim.x + threadIdx.x;
  if (e < E) atomicAdd(&counts[etype[e]], 1);
}

// ---------------- kernel: prefix + tile table (single thread; R=200) ----------------
__global__ void k_scan(const int* __restrict__ counts, int* cursor,
                       int* tileRel, int* tileStart, int* tileEnd,
                       int* numTiles, int R) {
  if (threadIdx.x == 0 && blockIdx.x == 0) {
    int eoff = 0, toff = 0;
    for (int r = 0; r < R; ++r) {
      cursor[r] = eoff;
      int c = counts[r];
      int nt = (c + TM - 1) / TM;
      for (int t = 0; t < nt; ++t) {
        tileRel[toff + t]   = r;
        tileStart[toff + t] = eoff + t * TM;
        tileEnd[toff + t]   = eoff + c;     // segment end for masking
      }
      toff += nt;
      eoff += c;
    }
    numTiles[0] = toff;
  }
}

// ---------------- kernel: scatter edges into relation-sorted order + degree ----------------
__global__ void k_scatter(const int* __restrict__ etype,
                          const int* __restrict__ dst,
                          int* cursor, int* perm, float* deg, int E) {
  int e = blockIdx.x * blockDim.x + threadIdx.x;
  if (e < E) {
    int pos = atomicAdd(&cursor[etype[e]], 1);
    perm[pos] = e;
    atomicAdd(&deg[dst[e]], 1.0f);
  }
}

// ---------------- kernel: relation-batched edge GEMM (f32 WMMA) ----------------
// Block: 256 threads (8 waves). Tile: 32 edges x 256 OUT, K=256.
// Wave w covers N-columns [16w,16w+16) and [16w+128,16w+144); each B fragment
// feeds two WMMAs (M rows 0-15 and 16-31) for 2x L2 W-traffic reuse.
__global__ void k_edge_gemm(const float* __restrict__ in_feat,
                            const int*   __restrict__ src,
                            const int*   __restrict__ dst,
                            const int*   __restrict__ perm,
                            const int*   __restrict__ tileRel,
                            const int*   __restrict__ tileStart,
                            const int*   __restrict__ tileEnd,
                            const int*   __restrict__ numTiles,
                            const float* __restrict__ Wg,
                            float* __restrict__ agg) {
  __shared__ float Ald[TM][LDP];
  __shared__ int   sSrc[TM];
  __shared__ int   sDst[TM];

  int tile = blockIdx.x;
  if (tile >= numTiles[0]) return;

  int r      = tileRel[tile];
  int start  = tileStart[tile];
  int segEnd = tileEnd[tile];

  if ((int)threadIdx.x < TM) {
    int gp = start + (int)threadIdx.x;
    int e  = (gp < segEnd) ? perm[gp] : -1;
    sSrc[threadIdx.x] = (e >= 0) ? src[e] : -1;
    sDst[threadIdx.x] = (e >= 0) ? dst[e] : -1;
  }
  __syncthreads();

  // stage gathered A rows (h_src) into LDS with 128-bit loads; in_feat is L2-resident
  for (int i = threadIdx.x; i < TM * (IN / 4); i += blockDim.x) {
    int m  = i >> 6;            // IN/4 == 64
    int c4 = i & 63;
    int s  = sSrc[m];
    float4 v = make_float4(0.f, 0.f, 0.f, 0.f);
    if (s >= 0) v = ((const float4*)(in_feat + (size_t)s * IN))[c4];
    *(float4*)&Ald[m][c4 * 4] = v;
  }
  __syncthreads();

  int lane = threadIdx.x & 31;
  int wave = threadIdx.x >> 5;
  int hs   = lane >> 4;        // lanes 16-31 hold K+2,K+3 (A) / rows K+2,K+3 (B)
  int l15  = lane & 15;

  const float* Wr = Wg + (size_t)r * IN * OUT;   // L2-resident (51 MB total)
  __builtin_prefetch(Wr, 0, 1);

  v8f acc00 = {}, acc01 = {};   // M-tile 0 (rows 0-15), N-tiles n0/n1
  v8f acc10 = {}, acc11 = {};   // M-tile 1 (rows 16-31)
  int n0 = wave * 16;
  int n1 = wave * 16 + 128;

  for (int k0 = 0; k0 < IN; k0 += 4) {
    int ka = k0 + hs * 2;
    v2f a0, a1;
    a0.x = Ald[l15][ka];        a0.y = Ald[l15][ka + 1];
    a1.x = Ald[16 + l15][ka];   a1.y = Ald[16 + l15][ka + 1];
    v2f b0, b1;
    b0.x = Wr[(size_t)ka * OUT + n0 + l15];
    b0.y = Wr[(size_t)(ka + 1) * OUT + n0 + l15];
    b1.x = Wr[(size_t)ka * OUT + n1 + l15];
    b1.y = Wr[(size_t)(ka + 1) * OUT + n1 + l15];
    acc00 = __builtin_amdgcn_wmma_f32_16x16x4_f32(false, a0, false, b0,
                                                  (short)0, acc00, false, false);
    acc01 = __builtin_amdgcn_wmma_f32_16x16x4_f32(false, a0, false, b1,
                                                  (short)0, acc01, false, false);
    acc10 = __builtin_amdgcn_wmma_f32_16x16x4_f32(false, a1, false, b0,
                                                  (short)0, acc10, false, false);
    acc11 = __builtin_amdgcn_wmma_f32_16x16x4_f32(false, a1, false, b1,
                                                  (short)0, acc11, false, false);
  }

  // C/D layout: VGPR j -> M = j (lanes 0-15) or j+8 (lanes 16-31), N = lane&15
#pragma unroll
  for (int j = 0; j < 8; ++j) {
    int m0 = j + hs * 8;
    int d0 = sDst[m0];
    if (d0 >= 0) {
      size_t base = (size_t)d0 * OUT;
      atomicAdd(&agg[base + n0 + l15], acc00[j]);
      atomicAdd(&agg[base + n1 + l15], acc01[j]);
    }
    int d1 = sDst[16 + m0];
    if (d1 >= 0) {
      size_t base = (size_t)d1 * OUT;
      atomicAdd(&agg[base + n0 + l15], acc10[j]);
      atomicAdd(&agg[base + n1 + l15], acc11[j]);
    }
  }
}

// ---------------- kernel: self-loop GEMM + mean epilogue (f32 WMMA) ----------------
__global__ void k_final(const float* __restrict__ in_feat,
                        const float* __restrict__ Wsl,
                        const float* __restrict__ agg,
                        const float* __restrict__ deg,
                        float* __restrict__ out, int N) {
  __shared__ float Ald[TM][LDP];
  int node0 = blockIdx.x * TM;

  for (int i = threadIdx.x; i < TM * (IN / 4); i += blockDim.x) {
    int m  = i >> 6;
    int c4 = i & 63;
    int node = node0 + m;
    float4 v = make_float4(0.f, 0.f, 0.f, 0.f);
    if (node < N) v = ((const float4*)(in_feat + (size_t)node * IN))[c4];
    *(float4*)&Ald[m][c4 * 4] = v;
  }
  __syncthreads();

  int lane = threadIdx.x & 31;
  int wave = threadIdx.x >> 5;
  int hs   = lane >> 4;
  int l15  = lane & 15;

  v8f acc00 = {}, acc01 = {};
  v8f acc10 = {}, acc11 = {};
  int n0 = wave * 16;
  int n1 = wave * 16 + 128;

  for (int k0 = 0; k0 < IN; k0 += 4) {
    int ka = k0 + hs * 2;
    v2f a0, a1;
    a0.x = Ald[l15][ka];        a0.y = Ald[l15][ka + 1];
    a1.x = Ald[16 + l15][ka];   a1.y = Ald[16 + l15][ka + 1];
    v2f b0, b1;
    b0.x = Wsl[(size_t)ka * OUT + n0 + l15];
    b0.y = Wsl[(size_t)(ka + 1) * OUT + n0 + l15];
    b1.x = Wsl[(size_t)ka * OUT + n1 + l15];
    b1.y = Wsl[(size_t)(ka + 1) * OUT + n1 + l15];
    acc00 = __builtin_amdgcn_wmma_f32_16x16x4_f32(false, a0, false, b0,
                                                  (short)0, acc00, false, false);
    acc01 = __builtin_amdgcn_wmma_f32_16x16x4_f32(false, a0, false, b1,
                                                  (short)0, acc01, false, false);
    acc10 = __builtin_amdgcn_wmma_f32_16x16x4_f32(false, a1, false, b0,
                                                  (short)0, acc10, false, false);
    acc11 = __builtin_amdgcn_wmma_f32_16x16x4_f32(false, a1, false, b1,
                                                  (short)0, acc11, false, false);
  }

#pragma unroll
  for (int j = 0; j < 8; ++j) {
    int m0 = j + hs * 8;
    int node = node0 + m0;
    if (node < N) {
      float dinv = 1.0f / fmaxf(deg[node], 1.0f);
      size_t base = (size_t)node * OUT;
      out[base + n0 + l15] = acc00[j] + agg[base + n0 + l15] * dinv;
      out[base + n1 + l15] = acc01[j] + agg[base + n1 + l15] * dinv;
    }
    int node1 = node0 + 16 + m0;
    if (node1 < N) {
      float dinv = 1.0f / fmaxf(deg[node1], 1.0f);
      size_t base = (size_t)node1 * OUT;
      out[base + n0 + l15] = acc10[j] + agg[base + n0 + l15] * dinv;
      out[base + n1 + l15] = acc11[j] + agg[base + n1 + l15] * dinv;
    }
  }
}

// ---------------- launch ----------------
extern "C" void kernel_launch(void* const* d_in, const int* in_sizes, int n_in,
                              void* d_out, int out_size, void* d_ws, size_t ws_size,
                              hipStream_t stream) {
  const float* in_feat = (const float*)d_in[0];
  const float* basis   = (const float*)d_in[1];
  const float* coef    = (const float*)d_in[2];
  const float* slw     = (const float*)d_in[3];
  const int*   src     = (const int*)d_in[4];
  const int*   dst     = (const int*)d_in[5];
  const int*   etype   = (const int*)d_in[6];
  float* out = (float*)d_out;

  const int N = in_sizes[0] / IN;
  const int B = in_sizes[1] / (IN * OUT);
  const int R = in_sizes[2] / B;
  const int E = in_sizes[4];
  const int maxTiles = (E + TM - 1) / TM + R;

  // carve workspace (256B-aligned slabs)
  uintptr_t base = (uintptr_t)d_ws;
  size_t cur = 0;
  auto take = [&](size_t bytes) -> void* {
    void* p = (void*)(base + cur);
    cur += (bytes + 255) & ~(size_t)255;
    return p;
  };
  float* Wg        = (float*)take((size_t)R * IN * OUT * sizeof(float)); // 52 MB
  float* agg       = (float*)take((size_t)N * OUT * sizeof(float));      // 102 MB
  float* deg       = (float*)take((size_t)N * sizeof(float));
  int*   counts    = (int*)take((size_t)R * sizeof(int));
  int*   cursor    = (int*)take((size_t)R * sizeof(int));
  int*   perm      = (int*)take((size_t)E * sizeof(int));
  int*   tileRel   = (int*)take((size_t)maxTiles * sizeof(int));
  int*   tileStart = (int*)take((size_t)maxTiles * sizeof(int));
  int*   tileEnd   = (int*)take((size_t)maxTiles * sizeof(int));
  int*   numTiles  = (int*)take(256);
  (void)ws_size; (void)n_in; (void)out_size;

  k_zero<<<4096, 256, 0, stream>>>(agg, deg, counts, N, R);
  k_weights<<<dim3((IN * OUT) / 1024, R), 256, 0, stream>>>(coef, basis, Wg, B);
  k_hist<<<(E + 255) / 256, 256, 0, stream>>>(etype, counts, E);
  k_scan<<<1, 32, 0, stream>>>(counts, cursor, tileRel, tileStart, tileEnd,
                               numTiles, R);
  k_scatter<<<(E + 255) / 256, 256, 0, stream>>>(etype, dst, cursor, perm, deg, E);
  k_edge_gemm<<<maxTiles, 256, 0, stream>>>(in_feat, src, dst, perm, tileRel,
                                            tileStart, tileEnd, numTiles, Wg, agg);
  k_final<<<(N + TM - 1) / TM, 256, 0, stream>>>(in_feat, slw, agg, deg, out, N);
}